// BoostedNeuralLDPCDecoder_8718783611089
// MI455X (gfx1250) — compile-verified
//
#include <hip/hip_runtime.h>
#include <stdint.h>

// LDPC min-sum decoder, fused persistent-LDS design for MI455X (gfx1250).
// One workgroup per batch element b; all 10 iterations LDS-resident.
// Messages are 5-bit quantized -> stored as int8 (units of q_step = 2^-4).
// Per-b state: Z*EP = 384*324 = 121.5 KB int8 in LDS.
// CDNA5-specific: TDM tensor_load_to_lds prefetch of xa rows + s_wait_tensorcnt.

#define B_     64
#define N_     68
#define Z_     384
#define M_     46
#define DC_    7
#define E_     (M_ * DC_)    // 322
#define EP_    324           // padded row stride: 81 dwords, odd -> bank-conflict free
#define ITERS_ 10
#define QS     0.0625f       // q_step = 2^(1-5)
#define STAGE_ROWS 8
#define NTHREADS 768         // 24 wave32s; 2 threads per z lane

typedef uint32_t u32x4 __attribute__((ext_vector_type(4)));
typedef int32_t  i32x4 __attribute__((ext_vector_type(4)));
typedef int32_t  i32x8 __attribute__((ext_vector_type(8)));

__global__ __launch_bounds__(NTHREADS) void ldpc_fused_kernel(
    const float* __restrict__ xa,      // (B, N, Z)
    const float* __restrict__ cnw,     // (ITERS)
    const int*   __restrict__ e2v,     // (E)
    const int*   __restrict__ shift,   // (E)
    float*       __restrict__ out)     // (ITERS, B, Z, N)
{
  __shared__ __align__(16) signed char sPrev[Z_ * EP_];        // 124416 B
  __shared__ __align__(16) float       sStage[STAGE_ROWS * Z_]; // 12288 B
  __shared__ short sVarEdges[E_];
  __shared__ short sVarStart[N_ + 1];
  __shared__ short sShift[E_];
  __shared__ int   sCnt[N_];
  __shared__ float sW[ITERS_];

  const int tid  = threadIdx.x;
  const int zl   = tid % Z_;           // z lane 0..383
  const int half = tid / Z_;           // 0 or 1: splits n-loop / m-loop
  const int b = blockIdx.x;
  const float* __restrict__ xab = xa + (size_t)b * N_ * Z_;

  // ---------- TDM: async prefetch of xa[b, 0:STAGE_ROWS, :] into LDS ----------
#if defined(__AMDGCN__) && __has_builtin(__builtin_amdgcn_tensor_load_to_lds) && __has_builtin(__builtin_amdgcn_s_wait_tensorcnt)
  if (tid == 0) {
    uint64_t ga = (uint64_t)(uintptr_t)xab;
    uint32_t lds_off = (uint32_t)(uintptr_t)(void*)sStage; // low 32b of generic = LDS byte addr
    u32x4 g0 = {0, 0, 0, 0};
    i32x8 g1 = {0, 0, 0, 0, 0, 0, 0, 0};
    i32x4 g2 = {0, 0, 0, 0};
    i32x4 g3 = {0, 0, 0, 0};
    i32x8 g4 = {0, 0, 0, 0, 0, 0, 0, 0};
    g0[0] = 1u;                                   // count=1 valid descriptor
    g0[1] = lds_off;                              // lds_addr (bytes)
    g0[2] = (uint32_t)(ga & 0xffffffffu);         // global_addr[31:0]
    g0[3] = (uint32_t)((ga >> 32) & 0x01ffffffu)  // global_addr[56:32]
          | 0x80000000u;                          // type=2 ("image")
    g1[0] = 0x00020000;                           // data_size=2 (4 bytes)
    g1[1] = (int)((uint32_t)Z_ << 16);            // tensor_dim0[15:0]  @bits[63:48]
    g1[2] = (int)(((uint32_t)Z_ >> 16) | ((uint32_t)N_ << 16)); // dim0 hi | tensor_dim1 lo
    g1[3] = (int)(((uint32_t)N_ >> 16) | ((uint32_t)Z_ << 16)); // dim1 hi | tile_dim0 = Z
    g1[4] = (int)STAGE_ROWS;                      // tile_dim1 @bits[143:128]
    g1[5] = (int)Z_;                              // tensor_dim0_stride lo (rows Z apart)
    __builtin_amdgcn_tensor_load_to_lds(g0, g1, g2, g3, g4, 0);
    __builtin_amdgcn_s_wait_tensorcnt(0);
  }
#else
  for (int i = tid; i < STAGE_ROWS * Z_; i += NTHREADS) sStage[i] = xab[i];
#endif

  // ---------- build var->edge CSR + copy shifts/weights into LDS ----------
  if (tid < N_)     sCnt[tid] = 0;
  if (tid < ITERS_) sW[tid]   = cnw[tid];
  __syncthreads();
  if (tid < E_) {
    sShift[tid] = (short)shift[tid];
    atomicAdd(&sCnt[e2v[tid]], 1);
  }
  __syncthreads();
  if (tid == 0) {
    int run = 0;
    for (int n = 0; n < N_; ++n) { int c = sCnt[n]; sVarStart[n] = (short)run; sCnt[n] = run; run += c; }
    sVarStart[N_] = (short)run;
  }
  __syncthreads();
  if (tid < E_) {
    int pos = atomicAdd(&sCnt[e2v[tid]], 1);   // order within a var is irrelevant
    sVarEdges[pos] = (short)tid;
  }
  // zero message state (prev = 0)
  {
    int* p32 = (int*)sPrev;
    for (int i = tid; i < (Z_ * EP_) / 4; i += NTHREADS) p32[i] = 0;
  }
  __syncthreads();

  const int nLo = half ? 34 : 0,  nHi = half ? N_ : 34;   // var split per thread pair
  const int mLo = half ? 23 : 0,  mHi = half ? M_ : 23;   // check split per thread pair
  const int row = zl * EP_;

  for (int it = 0; it < ITERS_; ++it) {
    const float wi = sW[it];

    // ===== variable-node phase (also emits out[it-1] = xa_in + sum(prev)) =====
    for (int n = nLo; n < nHi; ++n) {
      float x0v = (n < STAGE_ROWS) ? sStage[n * Z_ + zl] : xab[n * Z_ + zl];
      int s0 = sVarStart[n], s1 = sVarStart[n + 1];
      int acc = 0;
      for (int k = s0; k < s1; ++k) acc += sPrev[row + sVarEdges[k]];
      float base = x0v + (float)acc * QS;          // fl(x0 + vsum): matches jax order
      if (it > 0)
        out[(((size_t)(it - 1) * B_ + b) * Z_ + zl) * N_ + n] = base;
      for (int k = s0; k < s1; ++k) {
        int e = sVarEdges[k];
        int p = sPrev[row + e];
        float t = (base - (float)p * QS) * 16.0f;  // |.|/q_step; *16 exact (pow2)
        int q = (int)__builtin_rintf(t);           // round-half-even == jnp.round
        q = q > 31 ? 31 : (q < -31 ? -31 : q);
        sPrev[row + e] = (signed char)q;           // slot now holds x2 (check input)
      }
    }
    __syncthreads();

    // ===== check-node phase: min-sum; reads x2 and writes msg to SAME slot =====
    for (int m = mLo; m < mHi; ++m) {
      int a[DC_], idx[DC_], negm = 0, par = 0;
#pragma unroll
      for (int j = 0; j < DC_; ++j) {
        int e = m * DC_ + j;
        int zp = zl + (int)sShift[e]; if (zp >= Z_) zp -= Z_;
        idx[j] = zp * EP_ + e;
        int v = sPrev[idx[j]];
        int neg = v < 0;
        negm |= neg << j; par ^= neg;
        a[j] = neg ? -v : v;
      }
      int m1 = a[0];
#pragma unroll
      for (int j = 1; j < DC_; ++j) m1 = a[j] < m1 ? a[j] : m1;
      int cnt = 0, m2 = 63;
#pragma unroll
      for (int j = 0; j < DC_; ++j) {
        if (a[j] == m1) cnt++; else m2 = a[j] < m2 ? a[j] : m2;
      }
#pragma unroll
      for (int j = 0; j < DC_; ++j) {
        int extm = (a[j] == m1 && cnt == 1) ? m2 : m1;
        float prod = wi * ((float)extm * QS);      // fl(w * ext_min): matches jax
        int mag = (int)__builtin_rintf(prod * 16.0f);
        mag = mag > 31 ? 31 : mag;
        int neg = (par ^ (negm >> j)) & 1;         // tot_sign * edge_sign
        sPrev[idx[j]] = (signed char)(neg ? -mag : mag); // = prev_new at var slot
      }
    }
    __syncthreads();
  }

  // ===== epilogue: out[ITERS-1] = xa_in + sum(prev_new) =====
  for (int n = nLo; n < nHi; ++n) {
    float x0v = (n < STAGE_ROWS) ? sStage[n * Z_ + zl] : xab[n * Z_ + zl];
    int s0 = sVarStart[n], s1 = sVarStart[n + 1];
    int acc = 0;
    for (int k = s0; k < s1; ++k) acc += sPrev[row + sVarEdges[k]];
    out[(((size_t)(ITERS_ - 1) * B_ + b) * Z_ + zl) * N_ + n] = x0v + (float)acc * QS;
  }
}

extern "C" void kernel_launch(void* const* d_in, const int* in_sizes, int n_in,
                              void* d_out, int out_size, void* d_ws, size_t ws_size,
                              hipStream_t stream) {
  (void)in_sizes; (void)n_in; (void)d_ws; (void)ws_size; (void)out_size;
  const float* xa    = (const float*)d_in[0];
  const float* cnw   = (const float*)d_in[1];
  const int*   e2v   = (const int*)d_in[2];
  // d_in[3] = edge_to_check: by construction check m owns edges [7m, 7m+7) -> implicit
  const int*   shift = (const int*)d_in[4];
  ldpc_fused_kernel<<<dim3(B_), dim3(NTHREADS), 0, stream>>>(
      xa, cnw, e2v, shift, (float*)d_out);
}